// AdditiveAttention_1554778161227
// MI455X (gfx1250) — compile-verified
//
#include <hip/hip_runtime.h>
#include <hip/hip_bf16.h>

typedef float v2f __attribute__((ext_vector_type(2)));
typedef float v8f __attribute__((ext_vector_type(8)));

// ---- hardware tanh: CDNA5 has a TANH transcendental; avoid the OCML polynomial ----
__device__ __forceinline__ float fast_tanh(float x) {
#if __has_builtin(__builtin_amdgcn_tanhf)
    return __builtin_amdgcn_tanhf(x);
#else
    float y;
    asm("v_tanh_f32 %0, %1" : "=v"(y) : "v"(x));
    return y;
#endif
}

// ---- async global->LDS (CDNA5) with sync fallback ----
#if __has_builtin(__builtin_amdgcn_global_load_async_to_lds_b128) && \
    __has_builtin(__builtin_amdgcn_s_wait_asynccnt)
#define HAVE_ASYNC_LDS 1
typedef int vint4_t __attribute__((vector_size(16)));
typedef __attribute__((address_space(1))) vint4_t* async_gptr;
typedef __attribute__((address_space(3))) vint4_t* async_lptr;
#endif

#define BATCH 4
#define QLEN 256
#define KLEN 1024
#define IN_DIM 256      // Q_SIZE == K_SIZE == V_SIZE
#define HDIM 128
#define BQ 8            // query rows per block in fused kernel
#define KP_PAD 20       // 16 payload floats + 4 pad: keeps 16B alignment, spreads banks
#define NEGINF_SCORE (-1000000.0f)

// -----------------------------------------------------------------------------
// Projection: Y[m, n] = sum_i X[m, i] * W[n, i]   (X: [rows,256], W: [128,256])
// One wave per 16x16 tile; K-loop = 64 fp32 WMMAs.
// -----------------------------------------------------------------------------
__global__ __launch_bounds__(32) void proj_wmma(const float* __restrict__ X,
                                                const float* __restrict__ W,
                                                float* __restrict__ Y) {
    const int m0   = blockIdx.x * 16;
    const int n0   = blockIdx.y * 16;
    const int lane = threadIdx.x;
    const int l15  = lane & 15;
    const int half = lane >> 4;
    const int koff = half * 2;

    const float* xrow = X + (size_t)(m0 + l15) * IN_DIM + koff;
    const float* wrow = W + (size_t)(n0 + l15) * IN_DIM + koff;

    v8f c = {};
#pragma unroll 8
    for (int i = 0; i < IN_DIM; i += 4) {
        v2f a, bm;
        a[0]  = xrow[i];
        a[1]  = xrow[i + 1];
        bm[0] = wrow[i];
        bm[1] = wrow[i + 1];
        c = __builtin_amdgcn_wmma_f32_16x16x4_f32(false, a, false, bm,
                                                  (short)0, c, false, false);
    }
#pragma unroll
    for (int r = 0; r < 8; ++r)
        Y[(size_t)(m0 + half * 8 + r) * HDIM + n0 + l15] = c[r];
}

// -----------------------------------------------------------------------------
// Fused: scores = w_v . tanh(qp + kp), masked softmax over K, out = attn @ V.
// One block (256 thr = 8 waves) per (batch, 8-query tile). Grid = 4 * 32 = 128.
// -----------------------------------------------------------------------------
__global__ __launch_bounds__(256) void attn_fused(const float* __restrict__ qp,
                                                  const float* __restrict__ kp,
                                                  const float* __restrict__ V,
                                                  const int*   __restrict__ valid_lens,
                                                  const float* __restrict__ wv,
                                                  float* __restrict__ out) {
    __shared__ float s_q[BQ][HDIM];       //  4   KB
    __shared__ float s_wv[HDIM];          //  0.5 KB
    __shared__ float s_sc[BQ][KLEN];      // 32   KB (scores -> attn probs)
    __shared__ float s_k[256][KP_PAD];    // 20   KB (k-projection panel, 16 used)

    const int b   = blockIdx.x >> 5;
    const int q0  = (blockIdx.x & 31) * BQ;
    const int tid = threadIdx.x;

    {   // stage q-projection tile + w_v
        const float* qbase = qp + ((size_t)(b * QLEN + q0)) * HDIM;
        for (int i = tid; i < BQ * HDIM; i += 256)
            s_q[i >> 7][i & (HDIM - 1)] = qbase[i];
        if (tid < HDIM) s_wv[tid] = wv[tid];
    }
    __syncthreads();

    const int vl = valid_lens[b];
    const float* kpb = kp + (size_t)b * KLEN * HDIM;

    // ---- score stage: 4 waves of 256 k's; h staged in 16-wide LDS panels ----
    for (int kb = 0; kb < KLEN; kb += 256) {
        float acc[BQ];
#pragma unroll
        for (int q = 0; q < BQ; ++q) acc[q] = 0.0f;

        for (int hc = 0; hc < HDIM; hc += 16) {
            __syncthreads();   // previous panel consumers done
            // cooperative stage of kp[kb..kb+255][hc..hc+15] (1024 x 16B chunks)
#pragma unroll
            for (int i = 0; i < 4; ++i) {
                const int c   = tid + 256 * i;
                const int row = c >> 2;
                const int col = (c & 3) * 4;
                const float* src = kpb + (size_t)(kb + row) * HDIM + hc + col;
#if defined(HAVE_ASYNC_LDS)
                __builtin_amdgcn_global_load_async_to_lds_b128(
                    (async_gptr)src, (async_lptr)&s_k[row][col], 0, 0);
#else
                *(float4*)&s_k[row][col] = *(const float4*)src;
#endif
            }
#if defined(HAVE_ASYNC_LDS)
            __builtin_amdgcn_s_wait_asynccnt(0);
#endif
            __syncthreads();   // panel visible to all

            // this thread's k row is s_k[tid][*]
#pragma unroll
            for (int q = 0; q < BQ; ++q) {
                float a = 0.0f;
#pragma unroll
                for (int j = 0; j < 16; ++j)
                    a += s_wv[hc + j] * fast_tanh(s_q[q][hc + j] + s_k[tid][j]);
                acc[q] += a;
            }
        }
        const int  k  = kb + tid;
        const bool ok = (k < vl);
#pragma unroll
        for (int q = 0; q < BQ; ++q)
            s_sc[q][k] = ok ? acc[q] : NEGINF_SCORE;
    }
    __syncthreads();

    // ---- masked softmax over K: one wave32 per query row ----
    const int wvid = tid >> 5;
    const int lane = tid & 31;
    {
        const int q = wvid;  // 8 waves -> 8 rows
        float m = -3.402823466e38f;
        for (int k = lane; k < KLEN; k += 32) m = fmaxf(m, s_sc[q][k]);
#pragma unroll
        for (int off = 16; off; off >>= 1) m = fmaxf(m, __shfl_xor(m, off, 32));
        float s = 0.0f;
        for (int k = lane; k < KLEN; k += 32) {
            float e = __expf(s_sc[q][k] - m);
            s_sc[q][k] = e;
            s += e;
        }
#pragma unroll
        for (int off = 16; off; off >>= 1) s += __shfl_xor(s, off, 32);
        const float inv = 1.0f / s;
        for (int k = lane; k < KLEN; k += 32) s_sc[q][k] *= inv;
    }
    __syncthreads();

    // ---- out = attn @ V via fp32 WMMA (M padded 8->16 with zero rows) ----
    const int l15  = lane & 15;
    const int half = lane >> 4;
    const int koff = half * 2;
    const float* Vb = V + (size_t)b * KLEN * IN_DIM;

    for (int t = 0; t < 2; ++t) {
        const int n0 = (wvid * 2 + t) * 16;
        v8f c = {};
#pragma unroll 4
        for (int kc = 0; kc < KLEN; kc += 4) {
            v2f a, bm;
            a[0]  = (l15 < BQ) ? s_sc[l15][kc + koff]     : 0.0f;
            a[1]  = (l15 < BQ) ? s_sc[l15][kc + koff + 1] : 0.0f;
            bm[0] = Vb[(size_t)(kc + koff) * IN_DIM + n0 + l15];
            bm[1] = Vb[(size_t)(kc + koff + 1) * IN_DIM + n0 + l15];
            c = __builtin_amdgcn_wmma_f32_16x16x4_f32(false, a, false, bm,
                                                      (short)0, c, false, false);
        }
        if (half == 0) {
#pragma unroll
            for (int r = 0; r < BQ; ++r)
                out[((size_t)(b * QLEN) + q0 + r) * IN_DIM + n0 + l15] = c[r];
        }
    }
}

extern "C" void kernel_launch(void* const* d_in, const int* in_sizes, int n_in,
                              void* d_out, int out_size, void* d_ws, size_t ws_size,
                              hipStream_t stream) {
    (void)in_sizes; (void)n_in; (void)out_size; (void)ws_size;

    const float* queries    = (const float*)d_in[0];  // [4,256,256]
    const float* keys       = (const float*)d_in[1];  // [4,1024,256]
    const float* values     = (const float*)d_in[2];  // [4,1024,256]
    const int*   valid_lens = (const int*)  d_in[3];  // [4]
    const float* W_q        = (const float*)d_in[4];  // [128,256]
    const float* W_k        = (const float*)d_in[5];  // [128,256]
    const float* w_v        = (const float*)d_in[6];  // [128]
    float* out = (float*)d_out;                       // [4,256,256]

    float* qp = (float*)d_ws;                         // [4*256, 128]  512 KB
    float* kp = qp + (size_t)BATCH * QLEN * HDIM;     // [4*1024,128]  2 MB

    proj_wmma<<<dim3((BATCH * QLEN) / 16, HDIM / 16), 32, 0, stream>>>(queries, W_q, qp);
    proj_wmma<<<dim3((BATCH * KLEN) / 16, HDIM / 16), 32, 0, stream>>>(keys,    W_k, kp);

    attn_fused<<<BATCH * (QLEN / BQ), 256, 0, stream>>>(qp, kp, values, valid_lens, w_v, out);
}